// Encoder_61770219651232
// MI455X (gfx1250) — compile-verified
//
#include <hip/hip_runtime.h>
#include <math.h>

typedef __attribute__((ext_vector_type(2))) float v2f;
typedef __attribute__((ext_vector_type(4))) float f4;
typedef __attribute__((ext_vector_type(8))) float v8f;

#define N_NODES 2000000
#define NM 10   // N_MODES / 4 weights per node

// Each block: 256 threads = 8 wave32. Each wave owns 32 nodes:
//   - two 16-node WMMA tiles compute qs = W_tile(16x10) @ qm4(10x4)
//     via 3x V_WMMA_F32_16X16X4_F32 (K padded 10->12, N padded 4->16)
//   - C tiles transposed through a per-wave LDS slice
//   - then each lane normalizes its node's quaternion, builds R, rotates V.
__global__ __launch_bounds__(256) void dqs_kernel(
    const float* __restrict__ x,    // 40 = qm4 row-major (10 x 4)
    const float* __restrict__ W,    // N_NODES x 10
    const float* __restrict__ VR,   // N_NODES x 4
    float* __restrict__ out)        // N_NODES x 4
{
    __shared__ float qbuf[8][32][4];   // per-wave qs staging (4 KB)

    const int tid  = threadIdx.x;
    const int wave = tid >> 5;
    const int lane = tid & 31;
    const int col  = lane & 15;   // N index (A: M index)
    const int kh   = lane >> 4;   // K half: 0 -> K{0,1}, 1 -> K{2,3} within a chunk

    const int nodeBase = blockIdx.x * 256 + wave * 32;

    // ---- B fragments (4x16, K x N) from qm4 = x.reshape(10,4); shared by both tiles.
    // VGPR j of chunk c holds K = 4c + 2*kh + j for lane's N = col.
    // Loads use a clamped (always in-bounds) column; padding selected to 0 with cndmask.
    v2f B0, B1, B2;
    {
        const int  bc  = (col < 4) ? col : 3;    // clamped address column
        const bool on  = (col < 4);
        const bool on2 = on && (kh == 0);        // chunk 2: K=10,11 are padding
        const int  k0  = 2 * kh;
        const float b0x = x[(0 + k0) * 4 + bc];
        const float b0y = x[(1 + k0) * 4 + bc];
        const float b1x = x[(4 + k0) * 4 + bc];
        const float b1y = x[(5 + k0) * 4 + bc];
        const float b2x = x[32 + bc];            // K=8
        const float b2y = x[36 + bc];            // K=9
        B0.x = on  ? b0x : 0.0f;  B0.y = on  ? b0y : 0.0f;
        B1.x = on  ? b1x : 0.0f;  B1.y = on  ? b1y : 0.0f;
        B2.x = on2 ? b2x : 0.0f;  B2.y = on2 ? b2y : 0.0f;
    }

#pragma unroll
    for (int t = 0; t < 2; ++t) {
        // ---- A fragment (16x4 per chunk): lane's M = col, VGPR j + kh select K.
        int g = nodeBase + t * 16 + col;
        if (g > N_NODES - 1) g = N_NODES - 1;          // clamp: keep EXEC all-ones
        const float* __restrict__ row = W + (size_t)g * NM;
        const int k0 = 2 * kh;

        v2f A0, A1, A2;
        A0.x = row[k0 + 0];
        A0.y = row[k0 + 1];
        A1.x = row[k0 + 4];
        A1.y = row[k0 + 5];
        // Always load K=8,9 (in-bounds for every row), zero-select on kh==1 lanes
        // (their slots are the K=10,11 padding).
        const float a2x = row[8];
        const float a2y = row[9];
        A2.x = (kh == 0) ? a2x : 0.0f;
        A2.y = (kh == 0) ? a2y : 0.0f;

        v8f C = {};
        C = __builtin_amdgcn_wmma_f32_16x16x4_f32(false, A0, false, B0, (short)0, C, false, false);
        C = __builtin_amdgcn_wmma_f32_16x16x4_f32(false, A1, false, B1, (short)0, C, false, false);
        C = __builtin_amdgcn_wmma_f32_16x16x4_f32(false, A2, false, B2, (short)0, C, false, false);

        // C/D layout: lanes 0-15 -> M = j, lanes 16-31 -> M = j + 8; N = col.
        if (col < 4) {
#pragma unroll
            for (int j = 0; j < 8; ++j) {
                qbuf[wave][t * 16 + kh * 8 + j][col] = C[j];
            }
        }
    }

    __syncthreads();

    // ---- per-node epilogue: one node per lane.
    {
        const int g   = nodeBase + lane;
        const bool ok = (g < N_NODES);
        const int gc  = ok ? g : (N_NODES - 1);

        float qx = qbuf[wave][lane][0];
        float qy = qbuf[wave][lane][1];
        float qz = qbuf[wave][lane][2];
        float qw = qbuf[wave][lane][3];

        const float inv = 1.0f / sqrtf(qx * qx + qy * qy + qz * qz + qw * qw);
        qx *= inv; qy *= inv; qz *= inv; qw *= inv;

        const f4 v = *(const f4*)(VR + (size_t)gc * 4);

        const float xx = qx * qx, yy = qy * qy, zz = qz * qz, ww = qw * qw;
        const float xy = qx * qy, xz = qx * qz, xw = qx * qw;
        const float yz = qy * qz, yw = qy * qw, zw = qz * qw;

        const float r00 = ww + xx - yy - zz;
        const float r01 = 2.0f * (xy - zw);
        const float r02 = 2.0f * (xz + yw);
        const float r10 = 2.0f * (xy + zw);
        const float r11 = ww - xx + yy - zz;
        const float r12 = 2.0f * (yz - xw);
        const float r20 = 2.0f * (xz - yw);
        const float r21 = 2.0f * (xw + yz);
        const float r22 = ww - xx - yy + zz;

        f4 y;
        y.x = r00 * v.x + r01 * v.y + r02 * v.z;
        y.y = r10 * v.x + r11 * v.y + r12 * v.z;
        y.z = r20 * v.x + r21 * v.y + r22 * v.z;
        y.w = v.w;

        if (ok) *(f4*)(out + (size_t)g * 4) = y;
    }
}

extern "C" void kernel_launch(void* const* d_in, const int* in_sizes, int n_in,
                              void* d_out, int out_size, void* d_ws, size_t ws_size,
                              hipStream_t stream) {
    (void)in_sizes; (void)n_in; (void)out_size; (void)d_ws; (void)ws_size;
    const float* x  = (const float*)d_in[0];
    const float* W  = (const float*)d_in[1];
    const float* VR = (const float*)d_in[2];
    float* out      = (float*)d_out;

    const int blocks = (N_NODES + 255) / 256;   // 256 nodes per block
    dqs_kernel<<<blocks, 256, 0, stream>>>(x, W, VR, out);
}